// SimplestHandwritingPredictionModel_38096359916086
// MI455X (gfx1250) — compile-verified
//
#include <hip/hip_runtime.h>

#define UNITS   400
#define G4      1600      // 4*UNITS
#define FEAT    3
#define BATCH   128
#define TSTEPS  1000
#define BT      16        // batch rows per workgroup (one WMMA M-tile)
#define NWAVES  10
#define NTPW    10        // 16-wide N tiles per wave: 10*10*16 = 1600
#define NT_TOTAL 100      // total 16-wide N tiles over G4
#define KT      (UNITS/4) // 100 k-steps of 4
#define NTHREADS (NWAVES*32)
#define HS      404       // padded h/c row stride (bank-conflict-free A loads)
#define ZS      1604      // padded z row stride

// packed-U layout: ((tile*KT + kt)*32 + lane) * 2 floats
#define PK_FLOATS ((size_t)NT_TOTAL * KT * 32 * 2)   // 640,000 floats = 2.56 MB

typedef __attribute__((ext_vector_type(2))) float v2f;
typedef __attribute__((ext_vector_type(8))) float v8f;

__device__ __forceinline__ float sigmoid_f(float x) {
    return 1.0f / (1.0f + __expf(-x));   // v_exp_f32 trans op
}

__device__ __forceinline__ float tanh_f(float x) {
#if __has_builtin(__builtin_amdgcn_tanhf)
    return __builtin_amdgcn_tanhf(x);    // v_tanh_f32 trans op on CDNA5
#else
    return 1.0f - 2.0f / (__expf(2.0f * x) + 1.0f);
#endif
}

// One-time repack of U into WMMA B-fragment order: each lane's two B elements
// (rows k0+2*lk, k0+2*lk+1 at column tile*16+lm) become one contiguous float2,
// so the recurrent loop does a single coalesced global_load_b64 per fragment.
__global__ __launch_bounds__(256)
void repack_U(const float* __restrict__ U, float* __restrict__ Upk) {
    const int idx = blockIdx.x * blockDim.x + threadIdx.x;  // (tile*KT+kt)*32+lane
    if (idx >= NT_TOTAL * KT * 32) return;
    const int lane = idx & 31;
    const int kt   = (idx >> 5) % KT;
    const int tile = idx / (KT * 32);
    const int plm  = lane & 15;
    const int plk  = lane >> 4;
    const int col  = tile * 16 + plm;
    const int row  = kt * 4 + 2 * plk;
    v2f v;
    v.x = U[(size_t)row * G4 + col];
    v.y = U[(size_t)(row + 1) * G4 + col];
    *(v2f*)&Upk[(size_t)idx * 2] = v;
}

template <bool PACKED>
__global__ __launch_bounds__(NTHREADS)
void lstm_persist_wmma(const float* __restrict__ x,
                       const float* __restrict__ W,
                       const float* __restrict__ U,
                       const float* __restrict__ Upk,
                       const float* __restrict__ b,
                       const float* __restrict__ p_i,
                       const float* __restrict__ p_f,
                       const float* __restrict__ p_o,
                       const float* __restrict__ Wd,
                       const float* __restrict__ bd,
                       float* __restrict__ out) {
    extern __shared__ float smem[];
    float* lds_h   = smem;                   // BT*HS
    float* lds_c   = lds_h + BT * HS;        // BT*HS
    float* lds_z   = lds_c + BT * HS;        // BT*ZS
    float* lds_W   = lds_z + BT * ZS;        // FEAT*G4
    float* lds_b   = lds_W + FEAT * G4;      // G4
    float* lds_pi  = lds_b + G4;             // UNITS
    float* lds_pf  = lds_pi + UNITS;         // UNITS
    float* lds_po  = lds_pf + UNITS;         // UNITS
    float* lds_WdB = lds_po + UNITS;         // KT*32*2 fragment-order padded Wd

    const int tid  = threadIdx.x;
    const int lane = tid & 31;
    const int wv   = tid >> 5;       // wave 0..9
    const int lm   = lane & 15;      // row/col within 16-tile
    const int lk   = lane >> 4;      // half-wave selector for K
    const int b0   = blockIdx.x * BT;

    // ---- one-time init: zero state, stage weights in LDS ----
    for (int i = tid; i < BT * HS; i += NTHREADS) { lds_h[i] = 0.0f; lds_c[i] = 0.0f; }
    for (int i = tid; i < FEAT * G4; i += NTHREADS) lds_W[i] = W[i];
    for (int i = tid; i < G4; i += NTHREADS) lds_b[i] = b[i];
    for (int i = tid; i < UNITS; i += NTHREADS) {
        lds_pi[i] = p_i[i]; lds_pf[i] = p_f[i]; lds_po[i] = p_o[i];
    }
    // Wd zero-padded N=3 -> 16, pre-swizzled into B-fragment order
    for (int i = tid; i < KT * 32; i += NTHREADS) {
        const int kt2 = i >> 5, ln = i & 31;
        const int plm = ln & 15, plk = ln >> 4;
        const int row = kt2 * 4 + 2 * plk;
        lds_WdB[i * 2]     = (plm < FEAT) ? Wd[row * FEAT + plm]       : 0.0f;
        lds_WdB[i * 2 + 1] = (plm < FEAT) ? Wd[(row + 1) * FEAT + plm] : 0.0f;
    }
    __syncthreads();

    // Per-lane invariant bases into U for B fragments
    const float* Ubase  = U + lm + (size_t)(2 * lk) * G4;      // direct path
    const float* PKbase = Upk + (size_t)wv * NTPW * KT * 64 + lane * 2;  // packed path
    const int wvcol = wv * (NTPW * 16);

    // Gate-phase assignment: 16 rows x 400 units = 6400 = 320 threads x 20
    const int gm  = tid / 20;
    const int gu0 = (tid % 20) * 20;

    const float bdv = (lm < FEAT) ? bd[lm] : 0.0f;

    for (int t = 0; t < TSTEPS; ++t) {
        // ============ phase A: z = h @ U via V_WMMA_F32_16X16X4_F32 ============
        v8f acc[NTPW];
#pragma unroll
        for (int j = 0; j < NTPW; ++j)
            acc[j] = (v8f){0.f, 0.f, 0.f, 0.f, 0.f, 0.f, 0.f, 0.f};

        for (int kt = 0; kt < KT; ++kt) {
            v2f a = *(const v2f*)&lds_h[lm * HS + (kt * 4 + 2 * lk)];
            if constexpr (PACKED) {
                const float* pk = PKbase + kt * 64;
#pragma unroll
                for (int j = 0; j < NTPW; ++j) {
                    v2f bf = *(const v2f*)(pk + (size_t)j * (KT * 64));
                    acc[j] = __builtin_amdgcn_wmma_f32_16x16x4_f32(
                        false, a, false, bf, (short)0, acc[j], false, false);
                }
            } else {
                const float* up = Ubase + (size_t)(kt * 4) * G4 + wvcol;
#pragma unroll
                for (int j = 0; j < NTPW; ++j) {
                    v2f bf;
                    bf.x = up[j * 16];
                    bf.y = up[j * 16 + G4];
                    acc[j] = __builtin_amdgcn_wmma_f32_16x16x4_f32(
                        false, a, false, bf, (short)0, acc[j], false, false);
                }
            }
        }
        // Store D tiles to LDS z: element (M = r + 8*lk, N = tile*16 + lm)
#pragma unroll
        for (int j = 0; j < NTPW; ++j) {
            const int col = wvcol + j * 16 + lm;
#pragma unroll
            for (int r = 0; r < 8; ++r)
                lds_z[(r + 8 * lk) * ZS + col] = acc[j][r];
        }
        __syncthreads();

        // ============ phase B: gates (fuses x@W + b + peepholes) ============
        {
            const float* xp = x + ((size_t)(b0 + gm) * TSTEPS + t) * FEAT;
            const float x0 = xp[0], x1 = xp[1], x2 = xp[2];
            for (int u = gu0; u < gu0 + 20; ++u) {
                const int ci = u, cf = UNITS + u, cg = 2 * UNITS + u, co = 3 * UNITS + u;
                float zi = lds_z[gm * ZS + ci] + lds_b[ci]
                         + x0 * lds_W[ci] + x1 * lds_W[G4 + ci] + x2 * lds_W[2 * G4 + ci];
                float zf = lds_z[gm * ZS + cf] + lds_b[cf]
                         + x0 * lds_W[cf] + x1 * lds_W[G4 + cf] + x2 * lds_W[2 * G4 + cf];
                float zg = lds_z[gm * ZS + cg] + lds_b[cg]
                         + x0 * lds_W[cg] + x1 * lds_W[G4 + cg] + x2 * lds_W[2 * G4 + cg];
                float zo = lds_z[gm * ZS + co] + lds_b[co]
                         + x0 * lds_W[co] + x1 * lds_W[G4 + co] + x2 * lds_W[2 * G4 + co];
                const float cprev = lds_c[gm * HS + u];
                const float ig = sigmoid_f(zi + lds_pi[u] * cprev);
                const float fg = sigmoid_f(zf + lds_pf[u] * cprev);
                const float cn = fg * cprev + ig * tanh_f(zg);
                const float og = sigmoid_f(zo + lds_po[u] * cn);
                const float hn = og * tanh_f(cn);
                lds_c[gm * HS + u] = cn;
                lds_h[gm * HS + u] = hn;
            }
        }
        __syncthreads();

        // ====== phase C: out[t] = h_t @ WdP + bd via WMMA on wave 0 ======
        // (only N columns 0..2 of the 16-wide padded tile are stored)
        if (wv == 0) {
            v8f pacc = (v8f){0.f, 0.f, 0.f, 0.f, 0.f, 0.f, 0.f, 0.f};
            for (int kt2 = 0; kt2 < KT; ++kt2) {
                v2f a  = *(const v2f*)&lds_h[lm * HS + (kt2 * 4 + 2 * lk)];
                v2f bf = *(const v2f*)&lds_WdB[(kt2 * 32 + lane) * 2];
                pacc = __builtin_amdgcn_wmma_f32_16x16x4_f32(
                    false, a, false, bf, (short)0, pacc, false, false);
            }
            if (lm < FEAT) {
#pragma unroll
                for (int r = 0; r < 8; ++r) {
                    const int m = r + 8 * lk;
                    out[((size_t)(b0 + m) * TSTEPS + t) * FEAT + lm] = pacc[r] + bdv;
                }
            }
        }
        // No barrier needed here: next phase A only *reads* lds_h, and the next
        // lds_h write (phase B) is fenced by the next __syncthreads().
    }
}

extern "C" void kernel_launch(void* const* d_in, const int* in_sizes, int n_in,
                              void* d_out, int out_size, void* d_ws, size_t ws_size,
                              hipStream_t stream) {
    (void)in_sizes; (void)n_in; (void)out_size;
    const float* x  = (const float*)d_in[0];
    const float* W  = (const float*)d_in[1];
    const float* U  = (const float*)d_in[2];
    const float* b  = (const float*)d_in[3];
    const float* pi = (const float*)d_in[4];
    const float* pf = (const float*)d_in[5];
    const float* po = (const float*)d_in[6];
    const float* Wd = (const float*)d_in[7];
    const float* bd = (const float*)d_in[8];
    float* out = (float*)d_out;

    const size_t lds_bytes =
        (size_t)(2 * BT * HS + BT * ZS + FEAT * G4 + G4 + 3 * UNITS + KT * 64)
        * sizeof(float);  // ~210 KB of the 320 KB WGP LDS

    if (ws_size >= PK_FLOATS * sizeof(float)) {
        float* Upk = (float*)d_ws;
        const int n = NT_TOTAL * KT * 32;
        hipLaunchKernelGGL(repack_U, dim3((n + 255) / 256), dim3(256), 0, stream, U, Upk);
        hipLaunchKernelGGL((lstm_persist_wmma<true>),
                           dim3(BATCH / BT), dim3(NTHREADS), lds_bytes, stream,
                           x, W, U, Upk, b, pi, pf, po, Wd, bd, out);
    } else {
        hipLaunchKernelGGL((lstm_persist_wmma<false>),
                           dim3(BATCH / BT), dim3(NTHREADS), lds_bytes, stream,
                           x, W, U, (const float*)nullptr, b, pi, pf, po, Wd, bd, out);
    }
}